// NGCF_31696858645080
// MI455X (gfx1250) — compile-verified
//
#include <hip/hip_runtime.h>

// ---------------- problem constants (from reference) ----------------
#define N_USER   50000
#define N_ITEM   100000
#define EMB      64
#define NNZ      1200000
#define BATCH    4096
#define N_NODES  (N_USER + N_ITEM)   // 150000
#define RATE     0.1f

typedef __attribute__((ext_vector_type(2))) float v2f;
typedef __attribute__((ext_vector_type(8))) float v8f;

// ---- sparse-dropout rescale of adjacency values (exact: floor(0.9+u)) ----
__global__ void prep_vals_kernel(const float* __restrict__ adj_vals,
                                 const float* __restrict__ drop_u,
                                 float* __restrict__ vals) {
  int g = blockIdx.x * blockDim.x + threadIdx.x;
  if (g < NNZ) {
    float m = floorf((1.0f - RATE) + drop_u[g]);   // 0/1 keep mask
    vals[g] = adj_vals[g] * m * (1.0f / (1.0f - RATE));
  }
}

// ---- ego = concat(user_emb, item_emb); running sum S starts as ego ----
__global__ void init_ego_kernel(const float* __restrict__ ue,
                                const float* __restrict__ ie,
                                float* __restrict__ A,
                                float* __restrict__ S) {
  int g = blockIdx.x * blockDim.x + threadIdx.x;   // exact grid: N_NODES*EMB
  float v = (g < N_USER * EMB) ? ue[g] : ie[g - N_USER * EMB];
  A[g] = v;
  S[g] = v;
}

// ---- edge-parallel SpMM with scatter-add: B[r,:] += v * A[c,:] ----
// 64 threads per edge; edge metadata broadcast within the two waves.
// Tables (38.4MB each) are L2-resident on MI455X (192MB L2) -> L2-atomic bound.
__global__ void spmm_kernel(const float* __restrict__ vals,
                            const int*   __restrict__ rows,
                            const int*   __restrict__ cols,
                            const float* __restrict__ A,
                            float*       __restrict__ B) {
  unsigned g = blockIdx.x * blockDim.x + threadIdx.x;  // exact grid: NNZ*EMB
  int e = (int)(g >> 6);
  int k = (int)(g & 63u);
  if (k == 0 && e + 2048 < NNZ) {
    // CDNA5 global_prefetch_b8: pull upcoming edge-list cachelines toward L2
    __builtin_prefetch(&cols[e + 2048], 0, 1);
    __builtin_prefetch(&vals[e + 2048], 0, 1);
    __builtin_prefetch(&rows[e + 2048], 0, 1);
  }
  float v = vals[e];
  if (v != 0.0f) {                       // ~10% dropped edges: skip atomics
    int c = cols[e];
    int r = rows[e];
    atomicAdd(&B[r * EMB + k], v * A[c * EMB + k]);
  }
}

// ---- S += H via the matrix pipe, bit-exact ----
// One wave per 16x16 f32 tile. D = sum_{k=0..3} I[:,4k:4k+4] x H[4k:4k+4,:] + C
// Each element gets exactly one 1.0*x product (exact) + C, so this equals a
// plain f32 add under RNE. Chained V_WMMA_F32_16X16X4_F32, accumulator in C.
// Grid sized exactly (37500 waves) -> EXEC all-ones as WMMA requires.
__global__ void wmma_accum_kernel(float* __restrict__ S,
                                  const float* __restrict__ H) {
  int wave = (blockIdx.x * blockDim.x + threadIdx.x) >> 5; // == tile id
  int lane = threadIdx.x & 31;
  int tileRow = wave >> 2;          // 9375 row-tiles of 16 nodes
  int tileCol = wave & 3;           // 4 column-tiles of 16 dims
  int row0 = tileRow * 16;
  int col0 = tileCol * 16;
  int m  = lane & 15;
  int hi = lane >> 4;

  // C/D layout: VGPR j holds (M = j + 8*hi, N = m)
  v8f c;
#pragma unroll
  for (int j = 0; j < 8; ++j)
    c[j] = S[(row0 + j + 8 * hi) * EMB + col0 + m];

#pragma unroll
  for (int k = 0; k < 4; ++k) {
    // A (16x4 f32): VGPR0 = K {0 | 2}, VGPR1 = K {1 | 3} by lane half.
    v2f a;
    a.x = (m == 4 * k + 2 * hi)     ? 1.0f : 0.0f;
    a.y = (m == 4 * k + 1 + 2 * hi) ? 1.0f : 0.0f;
    // B (4x16 f32): rows of H tile striped across lanes, same K split as A.
    v2f b;
    b.x = H[(row0 + 4 * k + 2 * hi)     * EMB + col0 + m];
    b.y = H[(row0 + 4 * k + 1 + 2 * hi) * EMB + col0 + m];
    c = __builtin_amdgcn_wmma_f32_16x16x4_f32(
        /*neg_a=*/false, a, /*neg_b=*/false, b,
        /*c_mod=*/(short)0, c, /*reuse_a=*/false, /*reuse_b=*/false);
  }

#pragma unroll
  for (int j = 0; j < 8; ++j)
    S[(row0 + j + 8 * hi) * EMB + col0 + m] = c[j];
}

// ---- gather outputs for batch rows whose level selector == `level` ----
// out = [u | p | n], each BATCH x EMB. Division (not reciprocal-mul) to match
// the reference's `running / float(i+1)` rounding exactly.
__global__ void gather_level_kernel(const float* __restrict__ S,
                                    const int* __restrict__ users,
                                    const int* __restrict__ pos,
                                    const int* __restrict__ neg,
                                    const int* __restrict__ ua,
                                    const int* __restrict__ pa,
                                    const int* __restrict__ na,
                                    const int* __restrict__ indexp,
                                    int level, float divisor,
                                    float* __restrict__ out) {
  int g = blockIdx.x * blockDim.x + threadIdx.x;  // exact grid: 3*BATCH*EMB
  int group = g / (BATCH * EMB);
  int rem   = g - group * (BATCH * EMB);
  int b = rem >> 6;
  int k = rem & 63;
  int idx0 = indexp[0];
  int a, node;
  if (group == 0)      { a = ua[b]; node = users[b] + (idx0 ? N_USER : 0); }
  else if (group == 1) { a = pa[b]; node = pos[b]   + (idx0 ? 0 : N_USER); }
  else                 { a = na[b]; node = neg[b]   + (idx0 ? 0 : N_USER); }
  if (a == level)
    out[g] = S[node * EMB + k] / divisor;
}

extern "C" void kernel_launch(void* const* d_in, const int* in_sizes, int n_in,
                              void* d_out, int out_size, void* d_ws, size_t ws_size,
                              hipStream_t stream) {
  (void)in_sizes; (void)n_in; (void)out_size; (void)ws_size;

  const float* user_emb = (const float*)d_in[0];
  const float* item_emb = (const float*)d_in[1];
  const float* adj_vals = (const float*)d_in[2];
  const float* drop_u   = (const float*)d_in[3];
  const int*   adj_rows = (const int*)d_in[4];
  const int*   adj_cols = (const int*)d_in[5];
  const int*   users    = (const int*)d_in[6];
  const int*   pos      = (const int*)d_in[7];
  const int*   neg      = (const int*)d_in[8];
  const int*   u_a      = (const int*)d_in[9];
  const int*   p_a      = (const int*)d_in[10];
  const int*   n_a      = (const int*)d_in[11];
  const int*   indexp   = (const int*)d_in[12];
  float* out = (float*)d_out;

  // workspace layout (all 256B aligned by construction)
  const size_t valsBytes = (size_t)NNZ * sizeof(float);            // 4.8 MB
  const size_t nodeBytes = (size_t)N_NODES * EMB * sizeof(float);  // 38.4 MB
  char* ws = (char*)d_ws;
  float* vals = (float*)(ws);
  float* bufA = (float*)(ws + valsBytes);
  float* bufB = (float*)(ws + valsBytes + nodeBytes);
  float* bufS = (float*)(ws + valsBytes + 2 * nodeBytes);

  prep_vals_kernel<<<(NNZ + 255) / 256, 256, 0, stream>>>(adj_vals, drop_u, vals);
  init_ego_kernel<<<(N_NODES * EMB) / 256, 256, 0, stream>>>(user_emb, item_emb,
                                                             bufA, bufS);
  // level 0: embed[0] = ego (divisor 1 is exact)
  gather_level_kernel<<<(3 * BATCH * EMB) / 256, 256, 0, stream>>>(
      bufS, users, pos, neg, u_a, p_a, n_a, indexp, 0, 1.0f, out);

  float* cur = bufA;
  float* nxt = bufB;
  for (int i = 1; i <= 4; ++i) {
    hipMemsetAsync(nxt, 0, nodeBytes, stream);
    spmm_kernel<<<(NNZ * EMB) / 256, 256, 0, stream>>>(vals, adj_rows, adj_cols,
                                                       cur, nxt);
    // S += hop_i on the matrix pipe (exact WMMA chain); 37500 waves exactly.
    wmma_accum_kernel<<<9375, 128, 0, stream>>>(bufS, nxt);
    gather_level_kernel<<<(3 * BATCH * EMB) / 256, 256, 0, stream>>>(
        bufS, users, pos, neg, u_a, p_a, n_a, indexp, i, (float)(i + 1), out);
    float* t = cur; cur = nxt; nxt = t;
  }
}